// ModelNew_70927089926387
// MI455X (gfx1250) — compile-verified
//
#include <hip/hip_runtime.h>
#include <stdint.h>

// Problem constants (reference: B=128, S=131072, fp32 x, bool mask)
#define B_ROWS  128
#define SEQ     131072
#define THREADS 256                    // 8 wave32s per block
#define ITEMS   16                     // elements per thread in scan pass
#define CHUNK   (THREADS * ITEMS)      // 4096 elements / 16KB per block
#define NCHUNK  (SEQ / CHUNK)          // 32 chunks per row

// CDNA5 async global->LDS path (gfx1250). Guarded so compile never breaks.
#if defined(__HIP_DEVICE_COMPILE__) && defined(__gfx1250__) && \
    __has_builtin(__builtin_amdgcn_global_load_async_to_lds_b128) && \
    __has_builtin(__builtin_amdgcn_s_wait_asynccnt)
#define USE_ASYNC_LDS 1
#else
#define USE_ASYNC_LDS 0
#endif

typedef float v4f __attribute__((ext_vector_type(4)));
typedef int   v4i __attribute__((ext_vector_type(4)));

__device__ __forceinline__ float wave_incl_scan(float v, int lane) {
  #pragma unroll
  for (int d = 1; d < 32; d <<= 1) {
    float n = __shfl_up(v, d, 32);
    if (lane >= d) v += n;
  }
  return v;
}

// -------- Pass 1: per-(row,chunk) masked sum -> partials[row][chunk] --------
__global__ void k_partial(const float* __restrict__ x,
                          const uint8_t* __restrict__ mask,
                          float* __restrict__ partials) {
  const int chunk = blockIdx.x;
  const int row   = blockIdx.y;
  const int t     = threadIdx.x;
  const size_t base = (size_t)row * SEQ + (size_t)chunk * CHUNK;

  const float4*   x4 = reinterpret_cast<const float4*>(x + base);
  const uint32_t* m4 = reinterpret_cast<const uint32_t*>(mask + base);

  float s = 0.f;
  #pragma unroll
  for (int i = 0; i < ITEMS / 4; ++i) {      // strided, fully coalesced
    const int idx = i * THREADS + t;
    const float4   v = x4[idx];
    const uint32_t m = m4[idx];
    s += v.x * (float)( m        & 1u);
    s += v.y * (float)((m >> 8)  & 1u);
    s += v.z * (float)((m >> 16) & 1u);
    s += v.w * (float)((m >> 24) & 1u);
  }
  #pragma unroll
  for (int o = 16; o > 0; o >>= 1) s += __shfl_down(s, o, 32);

  __shared__ float red[THREADS / 32];
  const int lane = t & 31, wid = t >> 5;
  if (lane == 0) red[wid] = s;
  __syncthreads();
  if (t == 0) {
    float tot = 0.f;
    #pragma unroll
    for (int w = 0; w < THREADS / 32; ++w) tot += red[w];
    partials[row * NCHUNK + chunk] = tot;
  }
}

// -------- Pass 2: exclusive scan of the 32 chunk sums per row (1 wave) ------
__global__ void k_scan_partials(const float* __restrict__ partials,
                                float* __restrict__ offsets) {
  const int row  = blockIdx.x;
  const int lane = threadIdx.x;              // 32 threads = one wave32
  const float v    = partials[row * NCHUNK + lane];
  const float incl = wave_incl_scan(v, lane);
  offsets[row * NCHUNK + lane] = incl - v;   // exclusive prefix
}

// -------- Pass 3: full local scan + chunk offset, write output --------------
__global__ void k_final(const float* __restrict__ x,
                        const uint8_t* __restrict__ mask,
                        const float* __restrict__ offsets,
                        float* __restrict__ out) {
  const int chunk = blockIdx.x;
  const int row   = blockIdx.y;
  const int t     = threadIdx.x;
  const int lane  = t & 31, wid = t >> 5;
  const size_t base = (size_t)row * SEQ + (size_t)chunk * CHUNK;

  float v[ITEMS];

#if USE_ASYNC_LDS
  __shared__ float    tile[CHUNK];           // 16 KB
  __shared__ uint32_t mtile[CHUNK / 4];      //  4 KB
  {
    const float*    gx = x + base;
    const uint32_t* gm = reinterpret_cast<const uint32_t*>(mask + base);
    #pragma unroll
    for (int i = 0; i < ITEMS / 4; ++i) {    // 4 x b128 per lane: 16 KB tile
      const int idx = (i * THREADS + t) * 4; // float index, 16B granule
      __builtin_amdgcn_global_load_async_to_lds_b128(
          (v4i*)(gx + idx), (v4i*)(tile + idx), 0, 0);
    }
    // mask tile: one b128 per lane covers 4 KB
    __builtin_amdgcn_global_load_async_to_lds_b128(
        (v4i*)(gm + t * 4), (v4i*)(mtile + t * 4), 0, 0);
    __builtin_amdgcn_s_wait_asynccnt(0);
    __syncthreads();
  }
  {
    const int e0 = t * ITEMS;                // contiguous 16 elems per thread
    #pragma unroll
    for (int i = 0; i < ITEMS / 4; ++i) {
      const uint32_t m = mtile[(e0 >> 2) + i];
      v[i * 4 + 0] = tile[e0 + i * 4 + 0] * (float)( m        & 1u);
      v[i * 4 + 1] = tile[e0 + i * 4 + 1] * (float)((m >> 8)  & 1u);
      v[i * 4 + 2] = tile[e0 + i * 4 + 2] * (float)((m >> 16) & 1u);
      v[i * 4 + 3] = tile[e0 + i * 4 + 3] * (float)((m >> 24) & 1u);
    }
  }
#else
  {
    const float4*   x4 = reinterpret_cast<const float4*>(x + base + (size_t)t * ITEMS);
    const uint32_t* m4 = reinterpret_cast<const uint32_t*>(mask + base + (size_t)t * ITEMS);
    #pragma unroll
    for (int i = 0; i < ITEMS / 4; ++i) {
      const float4   q = x4[i];
      const uint32_t m = m4[i];
      v[i * 4 + 0] = q.x * (float)( m        & 1u);
      v[i * 4 + 1] = q.y * (float)((m >> 8)  & 1u);
      v[i * 4 + 2] = q.z * (float)((m >> 16) & 1u);
      v[i * 4 + 3] = q.w * (float)((m >> 24) & 1u);
    }
  }
#endif

  // sequential inclusive scan of this thread's 16 elements
  float run = 0.f;
  #pragma unroll
  for (int j = 0; j < ITEMS; ++j) { run += v[j]; v[j] = run; }

  // wave32 scan of thread totals
  const float incl  = wave_incl_scan(run, lane);
  const float texcl = incl - run;

  __shared__ float wsum[THREADS / 32];
  if (lane == 31) wsum[wid] = incl;          // wave total
  __syncthreads();
  float wprefix = 0.f;
  for (int w = 0; w < wid; ++w) wprefix += wsum[w];

  const float off = offsets[row * NCHUNK + chunk] + wprefix + texcl;

  // non-temporal streaming stores: keep x/mask resident in the 192MB L2
  v4f* o4 = reinterpret_cast<v4f*>(out + base + (size_t)t * ITEMS);
  #pragma unroll
  for (int i = 0; i < ITEMS / 4; ++i) {
    v4f q;
    q.x = v[i * 4 + 0] + off;
    q.y = v[i * 4 + 1] + off;
    q.z = v[i * 4 + 2] + off;
    q.w = v[i * 4 + 3] + off;
    __builtin_nontemporal_store(q, o4 + i);
  }
}

extern "C" void kernel_launch(void* const* d_in, const int* in_sizes, int n_in,
                              void* d_out, int out_size, void* d_ws, size_t ws_size,
                              hipStream_t stream) {
  (void)in_sizes; (void)n_in; (void)out_size; (void)ws_size;
  const float*   x    = (const float*)d_in[0];
  const uint8_t* mask = (const uint8_t*)d_in[1];
  float*         out  = (float*)d_out;

  float* partials = (float*)d_ws;                   // 128*32 floats
  float* offsets  = partials + B_ROWS * NCHUNK;     // 128*32 floats (32KB total)

  dim3 grid(NCHUNK, B_ROWS);
  k_partial      <<<grid,   THREADS, 0, stream>>>(x, mask, partials);
  k_scan_partials<<<B_ROWS, 32,      0, stream>>>(partials, offsets);
  k_final        <<<grid,   THREADS, 0, stream>>>(x, mask, offsets, out);
}